// GDDetector_7232724927240
// MI455X (gfx1250) — compile-verified
//
#include <hip/hip_runtime.h>

typedef __attribute__((ext_vector_type(2))) float v2f;
typedef __attribute__((ext_vector_type(8))) float v8f;

#define BATCH 16384
#define MDIM 192
#define NDIM 64

__global__ __launch_bounds__(512) void gd_detector_kernel(
    const float* __restrict__ y,
    const float* __restrict__ H,
    const float* __restrict__ x,
    const float* __restrict__ step,
    const int* __restrict__ iters_p,
    float* __restrict__ out)
{
    // H tile for this batch: 192x64 f32 = 48KB. After the WMMA phase the same
    // region is reused (aliased) for the 64x64 HtH matrix.
    __shared__ float Hs[MDIM * NDIM];
    __shared__ float ys[MDIM];
    __shared__ float xs[NDIM];
    __shared__ float htys[NDIM];

    const int b    = blockIdx.x;
    const int tid  = threadIdx.x;
    const int lane = tid & 31;
    const int w    = tid >> 5;          // wave id 0..15

    // ---- Phase 1: async DMA H into LDS (global_load_async_to_lds_b128,
    // GVS mode: saddr = batch base, vaddr = byte offset, vdst = LDS addr).
    // LDS address = low 32 bits of the flat shared pointer (ISA 10.2).
    {
        const float*   Hg   = H + (size_t)b * (MDIM * NDIM);
        const unsigned lds0 = (unsigned)(uintptr_t)(void*)Hs;
        #pragma unroll
        for (int i = 0; i < (MDIM * NDIM / 4) / 512; ++i) {  // 6 x b128 per thread
            const unsigned off = (unsigned)(tid + i * 512) * 16u;
            const unsigned lds = lds0 + off;
            asm volatile("global_load_async_to_lds_b128 %0, %1, %2"
                         :: "v"(lds), "v"(off), "s"(Hg) : "memory");
        }
        if (tid < MDIM)               ys[tid]        = y[(size_t)b * MDIM + tid];
        else if (tid < MDIM + NDIM)   xs[tid - MDIM] = x[(size_t)b * NDIM + (tid - MDIM)];
    }
    asm volatile("s_wait_asynccnt 0" ::: "memory");   // this wave's DMAs done
    __syncthreads();

    // ---- Phase 2: Hty (threads 0..63) ----
    if (tid < NDIM) {
        float acc = 0.f;
        #pragma unroll 8
        for (int m = 0; m < MDIM; ++m)
            acc += Hs[m * NDIM + tid] * ys[m];
        htys[tid] = acc;
    }

    // ---- Phase 3: HtH = H^T * H via V_WMMA_F32_16X16X4_F32 ----
    // HtH is symmetric and the (i,j)/(j,i) sums round identically, so only the
    // 10 upper-triangle 16x16 tiles are computed (waves 0..9); mirrored later.
    int ti, tj;
    if      (w < 4) { ti = 0;      tj = w;      }
    else if (w < 7) { ti = 1;      tj = w - 3;  }
    else if (w < 9) { ti = 2;      tj = w - 5;  }
    else            { ti = 3;      tj = 3;      }
    ti <<= 4; tj <<= 4;

    const int r  = lane & 15;
    const int hi = lane >> 4;           // 0 or 1
    const int kk = hi << 1;             // 0 or 2

    v8f c = {};
    if (w < 10) {
        #pragma unroll 4
        for (int k0 = 0; k0 < MDIM; k0 += 4) {
            const float* pk = Hs + (k0 + kk) * NDIM;
            v2f a, bm;
            a.x  = pk[ti + r];
            a.y  = pk[NDIM + ti + r];
            bm.x = pk[tj + r];
            bm.y = pk[NDIM + tj + r];
            c = __builtin_amdgcn_wmma_f32_16x16x4_f32(
                    /*neg_a=*/false, a, /*neg_b=*/false, bm,
                    /*c_mod=*/(short)0, c, /*reuse_a=*/false, /*reuse_b=*/false);
        }
    }
    __syncthreads();                    // everyone done reading Hs

    // ---- Phase 4: spill C frags into HtH (aliases the H region), mirroring
    // off-diagonal tiles. C layout: vgpr v -> row ti+v (lanes 0-15) or ti+v+8
    // (lanes 16-31), col tj + lane%16.
    float* HtHs = Hs;                   // 64x64, stride NDIM
    if (w < 10) {
        const int row0 = ti + (hi << 3);
        #pragma unroll
        for (int v = 0; v < 8; ++v)
            HtHs[(row0 + v) * NDIM + tj + r] = c[v];
        if (ti != tj) {
            #pragma unroll
            for (int v = 0; v < 8; ++v)
                HtHs[(tj + r) * NDIM + row0 + v] = c[v];   // transpose mirror
        }
    }
    __syncthreads();

    // ---- Phase 5: GD iterations on wave 0 only (single-wave LDS ops are
    // program-ordered, so no barriers; lanes own rows l and l+32) ----
    if (tid >= 32) return;

    const float s2    = 2.0f * step[0];
    const int   iters = iters_p[0];
    const int   l0 = lane, l1 = lane + 32;
    const float h0 = htys[l0], h1 = htys[l1];
    float nx0 = xs[l0], nx1 = xs[l1];

    for (int it = 0; it < iters; ++it) {
        float a0 = 0.f, a1 = 0.f;
        #pragma unroll 8
        for (int j = 0; j < NDIM; ++j) {
            const float xj = xs[j];
            a0 += HtHs[l0 * NDIM + j] * xj;
            a1 += HtHs[l1 * NDIM + j] * xj;
        }
        nx0 = nx0 + s2 * (h0 - a0);
        nx1 = nx1 + s2 * (h1 - a1);
        xs[l0] = nx0;                   // writes follow all reads of this iter
        xs[l1] = nx1;
    }

    out[(size_t)b * NDIM + l0] = nx0;
    out[(size_t)b * NDIM + l1] = nx1;
}

extern "C" void kernel_launch(void* const* d_in, const int* in_sizes, int n_in,
                              void* d_out, int out_size, void* d_ws, size_t ws_size,
                              hipStream_t stream) {
    const float* y     = (const float*)d_in[0];
    const float* H     = (const float*)d_in[1];
    const float* x     = (const float*)d_in[2];
    const float* step  = (const float*)d_in[3];
    const int*   iters = (const int*)d_in[4];
    float*       out   = (float*)d_out;

    gd_detector_kernel<<<dim3(BATCH), dim3(512), 0, stream>>>(
        y, H, x, step, iters, out);
}